// Noise_Regressor_66821101191340
// MI455X (gfx1250) — compile-verified
//
#include <hip/hip_runtime.h>
#include <hip/hip_bf16.h>

typedef float v2f __attribute__((ext_vector_type(2)));
typedef float v8f __attribute__((ext_vector_type(8)));

#define SLEN 2048
#define DMODEL 512
#define NIMU 13
#define NCOL 117      // NUM_IMUS * NUM_PARAMS
#define LDP 128       // padded leading dim for P
#define NPAD 128      // padded N for WMMA tiles
#define CHUNK 128

// workspace layout (in floats)
#define XN_OFF  0
#define P_OFF   (SLEN * DMODEL)              // 2048*512
#define PRE_OFF (P_OFF + SLEN * LDP)         // + 2048*128
#define WC_OFF  (PRE_OFF + 8 * NIMU * SLEN)  // + 8*13*2048
// Wc: column-major padded weights [NPAD][DMODEL] (Wc[n*512+k])

// ---------------- LayerNorm: one block per row ----------------
__global__ void ln_kernel(const float* __restrict__ x,
                          const float* __restrict__ w,
                          const float* __restrict__ b,
                          float* __restrict__ xn) {
    const int row = blockIdx.x;
    const int tid = threadIdx.x;                 // 256 threads, 2 elems each
    const float* xr = x + row * DMODEL;
    float a0 = xr[tid];
    float a1 = xr[tid + 256];

    __shared__ float s_sum[256];
    __shared__ float s_sq[256];
    s_sum[tid] = a0 + a1;
    s_sq[tid]  = a0 * a0 + a1 * a1;
    __syncthreads();
    for (int off = 128; off > 0; off >>= 1) {
        if (tid < off) {
            s_sum[tid] += s_sum[tid + off];
            s_sq[tid]  += s_sq[tid + off];
        }
        __syncthreads();
    }
    const float mu  = s_sum[0] * (1.0f / DMODEL);
    const float var = s_sq[0] * (1.0f / DMODEL) - mu * mu;
    const float inv = rsqrtf(var + 1e-5f);
    xn[row * DMODEL + tid]       = (a0 - mu) * inv * w[tid]       + b[tid];
    xn[row * DMODEL + tid + 256] = (a1 - mu) * inv * w[tid + 256] + b[tid + 256];
}

// ---------------- pack W (512x117 row-major) -> Wc (128 cols x 512 k, col-major, zero-padded) ----------------
__global__ void pack_w(const float* __restrict__ W, float* __restrict__ Wc) {
    const int idx = blockIdx.x * 256 + threadIdx.x;   // 0 .. 128*512-1
    const int n = idx >> 9;          // 0..127
    const int k = idx & 511;         // 0..511
    Wc[idx] = (n < NCOL) ? W[k * NCOL + n] : 0.0f;
}

// ---------------- GEMM via V_WMMA_F32_16X16X4_F32 ----------------
// One wave32 per 16x16 output tile. M tiles = 128, N tiles = 8 (117 padded to 128).
// A frag (16x4 f32): lanes 0-15 hold A[M=lane][k0+0..1]; lanes 16-31 hold A[M=lane-16][k0+2..3].
// B frag (4x16 f32) mirrored: lane half selects K pair, lane&15 selects N.
// With Wc column-major, both fragments are single b64 loads -> no divergence around WMMA.
// C/D (16x16 f32): VGPR v, lanes 0-15 -> M=v, lanes 16-31 -> M=v+8, N=lane&15.
__global__ void gemm_wmma(const float* __restrict__ Xn,
                          const float* __restrict__ Wc,
                          const float* __restrict__ bias,
                          float* __restrict__ P) {
    const int lane = threadIdx.x & 31;
    const int wid  = threadIdx.x >> 5;
    const int gw   = blockIdx.x * 8 + wid;   // 0..1023
    const int mt   = gw >> 3;                // 0..127
    const int nt   = gw & 7;                 // 0..7
    const int half = lane >> 4;              // 0 or 1
    const int l15  = lane & 15;

    const int arow = mt * 16 + l15;
    const int ncol = nt * 16 + l15;

    const float* aPtr = Xn + arow * DMODEL + half * 2;   // step +4 per iter
    const float* bPtr = Wc + ncol * DMODEL + half * 2;   // step +4 per iter

    v8f acc = {0.f, 0.f, 0.f, 0.f, 0.f, 0.f, 0.f, 0.f};

    #pragma unroll 4
    for (int k0 = 0; k0 < DMODEL; k0 += 4) {
        v2f a = *(const v2f*)(aPtr + k0);
        v2f bb = *(const v2f*)(bPtr + k0);
        acc = __builtin_amdgcn_wmma_f32_16x16x4_f32(
            /*neg_a=*/false, a, /*neg_b=*/false, bb,
            /*c_mod=*/(short)0, acc, /*reuse_a=*/false, /*reuse_b=*/false);
    }

    if (ncol < NCOL) {
        const float bv = bias[ncol];
        const int rbase = mt * 16 + half * 8;
        #pragma unroll
        for (int v = 0; v < 8; ++v) {
            P[(rbase + v) * LDP + ncol] = acc[v] + bv;
        }
    }
}

// ---------------- parameter extraction ----------------
__device__ __forceinline__ float softplusf(float x) {
    return (x > 20.0f) ? x : log1pf(__expf(x));
}

__global__ void prep_kernel(const float* __restrict__ P, float* __restrict__ pre) {
    const int idx = blockIdx.x * 256 + threadIdx.x;
    if (idx >= SLEN * NIMU) return;
    const int i = idx / NIMU;
    const int m = idx % NIMU;
    const float* pr = P + i * LDP;
    const float p0   = pr[0 * NIMU + m];
    const float p1   = pr[1 * NIMU + m];
    const float p2   = pr[2 * NIMU + m];
    const float p3   = pr[3 * NIMU + m];
    const float c    = pr[4 * NIMU + m];
    const float cth  = pr[5 * NIMU + m];
    const float phi  = pr[6 * NIMU + m];
    const float phth = pr[7 * NIMU + m];

    const float d    = sqrtf(p1 * p1 + 1e-5f);
    const float dth  = sqrtf(p3 * p3 + 1e-5f);
    // om = sqrt(4k - d^2)/2 with k = d^2/4 + softplus(p0)  =>  om = sqrt(softplus(p0))
    const float om   = sqrtf(softplusf(p0));
    const float omth = sqrtf(softplusf(p2));

    pre[(0 * NIMU + m) * SLEN + i] = c;
    pre[(1 * NIMU + m) * SLEN + i] = 0.5f * d;
    pre[(2 * NIMU + m) * SLEN + i] = om;
    pre[(3 * NIMU + m) * SLEN + i] = phi;
    pre[(4 * NIMU + m) * SLEN + i] = cth;
    pre[(5 * NIMU + m) * SLEN + i] = 0.5f * dth;
    pre[(6 * NIMU + m) * SLEN + i] = omth;
    pre[(7 * NIMU + m) * SLEN + i] = phth;
}

// ---------------- O(S^2 * NIMU) oscillator accumulation ----------------
// grid = (8 j-tiles, 13 imus), block = 256 threads (one j each).
__global__ void osc_kernel(const float* __restrict__ pre,
                           const float* __restrict__ P,
                           const float* __restrict__ minNorm,
                           float* __restrict__ out) {
    const int m = blockIdx.y;
    const int j = blockIdx.x * 256 + threadIdx.x;
    const float fj = (float)j;
    const int jTileMax = blockIdx.x * 256 + 255;

    __shared__ float s[8][CHUNK];
    float acc = 0.0f;

    for (int ic = 0; ic <= jTileMax; ic += CHUNK) {
        // cooperative stage of 8 param arrays for i in [ic, ic+CHUNK)
        for (int t = threadIdx.x; t < 8 * CHUNK; t += 256) {
            const int pp = t >> 7;           // / CHUNK
            const int ii = t & (CHUNK - 1);
            s[pp][ii] = pre[(pp * NIMU + m) * SLEN + ic + ii];
        }
        __syncthreads();

        const int nmax = j - ic;             // highest valid ii (may be < 0)
        const int lim  = nmax < (CHUNK - 1) ? nmax : (CHUNK - 1);
        for (int ii = 0; ii <= lim; ++ii) {
            const float ts = fj - (float)(ic + ii);
            const float e1 = __expf(-s[1][ii] * ts);
            const float s1 = __sinf(fmaf(ts, s[2][ii], s[3][ii]));
            const float e2 = __expf(-s[5][ii] * ts);
            const float s2 = __sinf(fmaf(ts, s[6][ii], s[7][ii]));
            acc = fmaf(s[0][ii] * e1, s1, acc);
            acc = fmaf(s[4][ii] * e2, s2, acc);
        }
        __syncthreads();
    }

    // out[m, j] = acc + min_orig_accel_norm + noise_bias[m, j] (= P[j, 8*13+m])
    out[m * SLEN + j] = acc + minNorm[0] + P[j * LDP + 8 * NIMU + m];
}

extern "C" void kernel_launch(void* const* d_in, const int* in_sizes, int n_in,
                              void* d_out, int out_size, void* d_ws, size_t ws_size,
                              hipStream_t stream) {
    const float* hs   = (const float*)d_in[0];   // (1, 2048, 512)
    const float* minN = (const float*)d_in[1];   // (1,)
    const float* lnw  = (const float*)d_in[2];   // (512,)
    const float* lnb  = (const float*)d_in[3];   // (512,)
    const float* W    = (const float*)d_in[4];   // (512, 117)
    const float* bv   = (const float*)d_in[5];   // (117,)

    float* ws  = (float*)d_ws;
    float* xn  = ws + XN_OFF;
    float* P   = ws + P_OFF;
    float* pre = ws + PRE_OFF;
    float* Wc  = ws + WC_OFF;
    float* out = (float*)d_out;                  // (13, 2048) fp32

    ln_kernel<<<SLEN, 256, 0, stream>>>(hs, lnw, lnb, xn);
    pack_w<<<(NPAD * DMODEL) / 256, 256, 0, stream>>>(W, Wc);
    gemm_wmma<<<128, 256, 0, stream>>>(xn, Wc, bv, P);
    prep_kernel<<<(SLEN * NIMU + 255) / 256, 256, 0, stream>>>(P, pre);
    osc_kernel<<<dim3(8, NIMU), 256, 0, stream>>>(pre, P, minN, out);
}